// FeatureVoxel3D_37941741093421
// MI455X (gfx1250) — compile-verified
//
#include <hip/hip_runtime.h>

// FeatureVoxel3D trilinear interpolation, MI455X (gfx1250, wave32).
//
// Mapping: 8 lanes per point, each lane owns a float4 channel quad.
//   - per corner: 8 lanes x 16B = one coalesced 128B cacheline (global_load_b128)
//   - out-of-range points (~42% of input) gather nothing (EXEC-masked loads),
//     store zeros.
//   - output stored with non-temporal hint to preserve L2 (192MB) for the
//     ~274MB voxel grid, which is the only data with reuse.

typedef __attribute__((ext_vector_type(4))) float v4f;

namespace {

constexpr int   RES = 128;   // cells per axis
constexpr int   DIM = 129;   // vertices per axis
constexpr int   CH  = 32;    // channels per vertex (= 128 bytes)

__global__ __launch_bounds__(256) void trilerp_voxel_kernel(
    const float* __restrict__ pos,    // [npts, 3]
    const float* __restrict__ vox,    // [129,129,129,32]
    float*       __restrict__ out,    // [npts, 32]
    int npts)
{
    const int gtid  = blockIdx.x * 256 + threadIdx.x;
    const int point = gtid >> 3;          // 8 lanes per point
    const int cq    = (gtid & 7) << 2;    // channel offset of this lane's float4
    if (point >= npts) return;

    // lo=0, hi=1 -> normalized p == raw position. 12B per point, lanes of the
    // same point hit the same cachelines (L0 absorbs the redundancy).
    const float px = pos[point * 3 + 0];
    const float py = pos[point * 3 + 1];
    const float pz = pos[point * 3 + 2];

    const bool ok = (px >= 0.f) & (px < 1.f) &
                    (py >= 0.f) & (py < 1.f) &
                    (pz >= 0.f) & (pz < 1.f);

    v4f acc = (v4f){0.f, 0.f, 0.f, 0.f};

    if (ok) {
        const float sx = px * (float)RES;
        const float sy = py * (float)RES;
        const float sz = pz * (float)RES;

        // truncation toward zero == jnp astype(int32); clamp is a no-op for
        // in-range points but matches the reference exactly.
        int ix = (int)sx; ix = ix < 0 ? 0 : (ix > RES - 1 ? RES - 1 : ix);
        int iy = (int)sy; iy = iy < 0 ? 0 : (iy > RES - 1 ? RES - 1 : iy);
        int iz = (int)sz; iz = iz < 0 ? 0 : (iz > RES - 1 ? RES - 1 : iz);

        const float fx = sx - (float)ix;
        const float fy = sy - (float)iy;
        const float fz = sz - (float)iz;
        const float gx = 1.f - fx, gy = 1.f - fy, gz = 1.f - fz;

        constexpr long SZ = CH;                   // +z: next vertex, 128B
        constexpr long SY = (long)DIM * CH;       // +y
        constexpr long SX = (long)DIM * DIM * CH; // +x

        const float* base =
            vox + (long)((ix * DIM + iy) * DIM + iz) * CH + cq;

        // Issue all 8 corner loads up front (8 outstanding LOADcnt),
        // then reduce. Each is a 16B-aligned b128 load.
        const v4f f000 = *(const v4f*)(base);
        const v4f f001 = *(const v4f*)(base + SZ);
        const v4f f010 = *(const v4f*)(base + SY);
        const v4f f011 = *(const v4f*)(base + SY + SZ);
        const v4f f100 = *(const v4f*)(base + SX);
        const v4f f101 = *(const v4f*)(base + SX + SZ);
        const v4f f110 = *(const v4f*)(base + SX + SY);
        const v4f f111 = *(const v4f*)(base + SX + SY + SZ);

        const float w000 = gx * gy * gz;
        const float w001 = gx * gy * fz;
        const float w010 = gx * fy * gz;
        const float w011 = gx * fy * fz;
        const float w100 = fx * gy * gz;
        const float w101 = fx * gy * fz;
        const float w110 = fx * fy * gz;
        const float w111 = fx * fy * fz;

        acc += w000 * f000;
        acc += w001 * f001;
        acc += w010 * f010;
        acc += w011 * f011;
        acc += w100 * f100;
        acc += w101 * f101;
        acc += w110 * f110;
        acc += w111 * f111;
    }

    // Streaming 128MB output: non-temporal store keeps L2 free for the grid.
    v4f* op = (v4f*)(out + (long)point * CH + cq);
    __builtin_nontemporal_store(acc, op);
}

} // namespace

extern "C" void kernel_launch(void* const* d_in, const int* in_sizes, int n_in,
                              void* d_out, int out_size, void* d_ws, size_t ws_size,
                              hipStream_t stream) {
    (void)n_in; (void)d_ws; (void)ws_size; (void)out_size;

    const float* positions = (const float*)d_in[0];  // [npts,3] f32
    const float* voxel     = (const float*)d_in[1];  // [129,129,129,32] f32
    float*       out       = (float*)d_out;          // [npts,32] f32

    const int npts = in_sizes[0] / 3;
    const long total_threads = (long)npts * 8;       // 8 lanes per point
    const int  block = 256;
    const int  grid  = (int)((total_threads + block - 1) / block);

    trilerp_voxel_kernel<<<grid, block, 0, stream>>>(positions, voxel, out, npts);
}